// DLMRI_65068754534724
// MI455X (gfx1250) — compile-verified
//
#include <hip/hip_runtime.h>
#include <math.h>

typedef __attribute__((ext_vector_type(2))) float v2f;
typedef __attribute__((ext_vector_type(8))) float v8f;

#define AS1 __attribute__((address_space(1)))
#define AS3 __attribute__((address_space(3)))
#if defined(__has_builtin)
#if __has_builtin(__builtin_amdgcn_global_load_async_to_lds_b32)
#define ASYNC_STAGE 1
#endif
#endif
#ifndef ASYNC_STAGE
#define ASYNC_STAGE 0
#endif

#define XDIM 80
#define YDIM 80
#define TDIM 12
#define NVOX (XDIM*YDIM*TDIM)        // 76800
#define NXP  83
#define NTP  15
#define NPOS (NXP*NXP*NTP)           // 103335 patch positions
#define ROWS (NPOS*2)                // 206670 FISTA rows (re/im channels)
#define TILES ((ROWS+15)/16)         // 12917 16-row WMMA tiles
#define KATOMS 128
#define DPATCH 64
#define ACCX 86
#define ACCT 18
#define NACC (2*ACCX*ACCX*ACCT)      // 266256 padded overlap-add accumulator

// ---- workspace layout (float offsets) ----
#define WS_XRE 0
#define WS_XIM (WS_XRE+NVOX)
#define WS_YRE (WS_XIM+NVOX)
#define WS_YIM (WS_YRE+NVOX)
#define WS_BRE (WS_YIM+NVOX)
#define WS_BIM (WS_BRE+NVOX)
#define WS_CRE (WS_BIM+NVOX)
#define WS_CIM (WS_CRE+NVOX)
#define WS_DTD (WS_CIM+NVOX)         // 128x128
#define WS_TWC (WS_DTD+16384)        // 80 twiddle cos
#define WS_TWS (WS_TWC+80)           // 80 twiddle sin
#define WS_COLS (WS_TWS+80)          // 128 column sums of D
#define WS_SCAL (WS_COLS+128)        // L, lam, thresh
#define WS_ACC (WS_SCAL+8)

// LDS partition for k_fista (floats)
#define LDS_DS    0                      // D padded: 64 rows, stride 132  (8448)
#define LDS_BFRAG 8448                   // DtD in B-fragment layout (16384)
#define LDS_ZS    (8448+16384)           // 4 waves x 16x132              (8448)
#define LDS_FLOATS (LDS_ZS + 4*2112)     // 33280 floats = 133120 bytes

// ------------------------------------------------------------------
__global__ void k_init(const float* __restrict__ y_ri, float* __restrict__ ws){
  int i = blockIdx.x*blockDim.x + threadIdx.x;
  if (i < NVOX){ ws[WS_YRE+i] = y_ri[2*i]; ws[WS_YIM+i] = y_ri[2*i+1]; }
  if (i < 80){
    float ang = 6.283185307179586f * (float)i / 80.0f;
    ws[WS_TWC+i] = cosf(ang); ws[WS_TWS+i] = sinf(ang);
  }
}

__global__ void k_zero(float* __restrict__ p, int n){
  int i = blockIdx.x*blockDim.x + threadIdx.x;
  if (i < n) p[i] = 0.0f;
}

// DtD = D^T D  (D is 64x128 row-major)
__global__ void k_dtd(const float* __restrict__ D, float* __restrict__ ws){
  int e = blockIdx.x*256 + threadIdx.x;
  if (e >= KATOMS*KATOMS) return;
  int i = e >> 7, j = e & 127;
  float s = 0.0f;
  for (int d = 0; d < DPATCH; ++d) s += D[d*KATOMS+i]*D[d*KATOMS+j];
  ws[WS_DTD+e] = s;
}

// power iteration for L, softplus scalars, colsum of D. 1 block x 128.
__global__ void k_power(const float* __restrict__ D, const float* __restrict__ lamp,
                        const float* __restrict__ betp, float* __restrict__ ws){
  __shared__ float b[128], v[128], nrm;
  int tid = threadIdx.x;
  const float* DtD = ws + WS_DTD;
  float cs = 0.0f;
  for (int d = 0; d < DPATCH; ++d) cs += D[d*KATOMS+tid];
  ws[WS_COLS+tid] = cs;
  b[tid] = 0.08838834764831845f;  // 1/sqrt(128)
  __syncthreads();
  for (int it = 0; it < 16; ++it){
    float s = 0.0f;
    for (int j = 0; j < 128; ++j) s += DtD[tid*128+j]*b[j];
    v[tid] = s; __syncthreads();
    if (tid == 0){ float n=0.0f; for (int j=0;j<128;++j) n += v[j]*v[j]; nrm = sqrtf(n); }
    __syncthreads();
    b[tid] = v[tid]/nrm; __syncthreads();
  }
  float s = 0.0f;
  for (int j = 0; j < 128; ++j) s += DtD[tid*128+j]*b[j];
  v[tid] = s; __syncthreads();
  if (tid == 0){
    float L = 0.0f; for (int j = 0; j < 128; ++j) L += b[j]*v[j];
    float lam  = log1pf(expf(lamp[0]));
    float beta = log1pf(expf(betp[0]));
    ws[WS_SCAL+0] = L; ws[WS_SCAL+1] = lam; ws[WS_SCAL+2] = beta/lam;
  }
}

// one 80-point DFT pass (ortho scaling). axis0 = X, axis1 = Y. sign=-1 fwd, +1 inv.
__global__ void k_dft(const float* __restrict__ ir, const float* __restrict__ ii,
                      float* __restrict__ outr, float* __restrict__ outi,
                      const float* __restrict__ twc, const float* __restrict__ tws,
                      int axis, float sign){
  __shared__ float sre[80], sim[80];
  int q = blockIdx.x; int a = q / TDIM, t = q % TDIM;
  int base, stride;
  if (axis == 0){ base = a*TDIM + t;       stride = YDIM*TDIM; }
  else          { base = a*YDIM*TDIM + t;  stride = TDIM; }
  int u = threadIdx.x;
  sre[u] = ir[base + u*stride]; sim[u] = ii[base + u*stride];
  __syncthreads();
  float cr = 0.0f, ci = 0.0f;
  for (int i = 0; i < 80; ++i){
    int k = (u*i) % 80;
    float c = twc[k], s = sign*tws[k];
    float xr = sre[i], xi = sim[i];
    cr += c*xr - s*xi;
    ci += c*xi + s*xr;
  }
  const float sc = 0.11180339887498948f;   // 1/sqrt(80)
  outr[base + u*stride] = cr*sc;
  outi[base + u*stride] = ci*sc;
}

__global__ void k_blend(float* __restrict__ kr, float* __restrict__ ki,
                        const float* __restrict__ yr, const float* __restrict__ yi,
                        const int* __restrict__ mask, const float* __restrict__ scal){
  int i = blockIdx.x*blockDim.x + threadIdx.x;
  if (i >= NVOX) return;
  if (mask[i]){
    float lam = scal[1]; float inv = 1.0f/(1.0f+lam);
    kr[i] = (kr[i] + lam*yr[i])*inv;
    ki[i] = (ki[i] + lam*yi[i])*inv;
  }
}

__global__ void k_crop(const float* __restrict__ acc, float* __restrict__ outr,
                       float* __restrict__ outi){
  int i = blockIdx.x*blockDim.x + threadIdx.x;
  if (i >= 2*NVOX) return;
  int ch = i / NVOX; int v = i % NVOX;
  int t = v % TDIM; int y = (v / TDIM) % YDIM; int x = v / (TDIM*YDIM);
  float val = acc[((ch*ACCX + x+3)*ACCX + y+3)*ACCT + t+3] * (1.0f/27.0f);
  if (ch == 0) outr[v] = val; else outi[v] = val;
}

__global__ void k_out(const float* __restrict__ xre, const float* __restrict__ xim,
                      float* __restrict__ out){
  int i = blockIdx.x*blockDim.x + threadIdx.x;
  if (i < NVOX){ out[2*i] = xre[i]; out[2*i+1] = xim[i]; }
}

// ------------------------------------------------------------------
// Fused per-tile FISTA: 16 rows/wave, all 8 iterations in-register.
// DtD is staged once per block into LDS, pre-swizzled into exact WMMA
// B-fragment layout -> the hot loop does ONE conflict-free ds_load_b64
// per v_wmma. D is staged (async-to-LDS when available) bank-padded for
// the one-shot xtD and gamma@D^T phases. z transposes via per-wave LDS.
__global__ __launch_bounds__(128, 1)
void k_fista(const float* __restrict__ xre, const float* __restrict__ xim,
             const float* __restrict__ D,   const float* __restrict__ DtD,
             const float* __restrict__ colsum, const float* __restrict__ scal,
             float* __restrict__ acc){
  extern __shared__ float smem[];
  float* Ds    = smem + LDS_DS;          // [64][132]
  v2f*   Bfrag = (v2f*)(smem + LDS_BFRAG); // 256 frags x 32 lanes
  const int lane = threadIdx.x & 31;
  const int wid  = threadIdx.x >> 5;
  const int g    = lane >> 4;
  const int m    = lane & 15;
  const int tile = blockIdx.x*4 + wid;
  float* zsw = smem + LDS_ZS + wid*2112; // [16][132] per wave

  // ---- stage D (async load-to-LDS path when the builtin exists) ----
  for (int i = threadIdx.x; i < DPATCH*KATOMS; i += 128){
    int d = i >> 7, k = i & 127;
#if ASYNC_STAGE
    __builtin_amdgcn_global_load_async_to_lds_b32(
        (AS1 int*)(unsigned long long)(D + i),
        (AS3 int*)(unsigned)(unsigned long long)(Ds + d*132 + k), 0, 0);
#else
    Ds[d*132 + k] = D[i];
#endif
  }
  // ---- stage DtD pre-swizzled into WMMA B-fragment records ----------
  for (int idx = threadIdx.x; idx < 256*32; idx += 128){
    int f = idx >> 5, l = idx & 31;
    int kb = f >> 3, j = f & 7;
    int gg = l >> 4, mm = l & 15;
    int rr = 4*kb + 2*gg;
    v2f val;
    val[0] = DtD[(rr  )*KATOMS + 16*j + mm];
    val[1] = DtD[(rr+1)*KATOMS + 16*j + mm];
    Bfrag[idx] = val;
  }
  for (int i = lane; i < 16*132; i += 32) zsw[i] = 0.0f;   // z = 0
#if ASYNC_STAGE
  asm volatile("s_wait_asynccnt 0x0" ::: "memory");
#endif
  __syncthreads();

  const float L    = scal[0];
  const float invL = 1.0f / L;
  const float sthr = scal[2] * invL;   // (beta/lam)/L

  // decode this lane's patch row (clamped for tail tiles)
  int rowm = tile*16 + m; if (rowm >= ROWS) rowm = ROWS-1;
  int ch = rowm & 1; int pp = rowm >> 1;
  int it = pp % NTP; pp /= NTP;
  int iy = pp % NXP;
  int ix = pp / NXP;
  const float* xsrc = ch ? xim : xre;
  __builtin_prefetch(xsrc + (ix*80 + iy)*12, 0, 1);

  // ---- xtd = (xp - mu) @ D, mean folded in via colsum -------------
  v8f xtd[8] = {};
  float lsum = 0.0f;
  for (int kb = 0; kb < 16; ++kb){
    int d0 = 4*kb + 2*g;
    v2f a;
#pragma unroll
    for (int vv = 0; vv < 2; ++vv){
      int d = d0 + vv;
      int pa = d >> 4, pb = (d >> 2) & 3, pc = d & 3;
      int sx = (ix + pa + 77) % 80;
      int sy = (iy + pb + 77) % 80;
      int st = (it + pc + 9)  % 12;
      a[vv] = xsrc[(sx*80 + sy)*12 + st];
    }
    lsum += a[0] + a[1];
#pragma unroll
    for (int j = 0; j < 8; ++j){
      v2f b;
      b[0] = Ds[(d0    )*132 + 16*j + m];
      b[1] = Ds[(d0 + 1)*132 + 16*j + m];
      xtd[j] = __builtin_amdgcn_wmma_f32_16x16x4_f32(false, a, false, b,
                                                     (short)0, xtd[j], false, false);
    }
  }
  float tot = lsum + __shfl_xor(lsum, 16, 32);
  float mu  = tot * (1.0f/64.0f);      // row mean, held by lanes m and m+16
#pragma unroll
  for (int j = 0; j < 8; ++j){
    float cs = colsum[16*j + m];
#pragma unroll
    for (int r = 0; r < 8; ++r){
      float muM = __shfl(mu, r + 8*g, 32);
      xtd[j][r] -= muM * cs;
    }
  }

  // ---- FISTA: 8 iterations, z in LDS (A-layout reload), gamma in regs
  v8f gam[8] = {};
  float t_f = 1.0f;
  for (int itF = 0; itF < 8; ++itF){
    v8f pz[8] = {};
    if (itF != 0){
      for (int kb = 0; kb < 32; ++kb){
        const v2f az = *(const v2f*)&zsw[m*132 + 4*kb + 2*g];
#pragma unroll
        for (int j = 0; j < 8; ++j){
          const v2f b = Bfrag[(kb*8 + j)*32 + lane];   // one ds_load_b64 / wmma
          pz[j] = __builtin_amdgcn_wmma_f32_16x16x4_f32(false, az, false, b,
                                                        (short)0, pz[j], false, false);
        }
      }
    }
    float tn   = 0.5f*(1.0f + sqrtf(1.0f + 4.0f*t_f*t_f));
    float coef = (t_f - 1.0f)/tn;
    t_f = tn;
#pragma unroll
    for (int j = 0; j < 8; ++j){
#pragma unroll
      for (int r = 0; r < 8; ++r){
        int zi = (r + 8*g)*132 + 16*j + m;
        float zc = zsw[zi];
        float u  = zc - (pz[j][r] - xtd[j][r])*invL;
        float au = fmaxf(fabsf(u) - sthr, 0.0f);
        float gn = copysignf(au, u);
        zsw[zi] = gn + coef*(gn - gam[j][r]);
        gam[j][r] = gn;
      }
    }
  }

  // ---- Dg = gamma @ D^T (+mu), then overlap-add scatter -----------
#pragma unroll
  for (int j = 0; j < 8; ++j)
#pragma unroll
    for (int r = 0; r < 8; ++r)
      zsw[(r + 8*g)*132 + 16*j + m] = gam[j][r];

  v8f dg[4] = {};
  for (int kb = 0; kb < 32; ++kb){
    const v2f ag = *(const v2f*)&zsw[m*132 + 4*kb + 2*g];
#pragma unroll
    for (int jb = 0; jb < 4; ++jb){
      const v2f b = *(const v2f*)&Ds[(16*jb + m)*132 + 4*kb + 2*g];
      dg[jb] = __builtin_amdgcn_wmma_f32_16x16x4_f32(false, ag, false, b,
                                                     (short)0, dg[jb], false, false);
    }
  }

  const int base = tile*16;
#pragma unroll
  for (int jb = 0; jb < 4; ++jb){
#pragma unroll
    for (int r = 0; r < 8; ++r){
      int M   = r + 8*g;
      float muM = __shfl(mu, M, 32);
      int row = base + M;
      float val = dg[jb][r] + muM;
      if (row < ROWS){
        int cch = row & 1; int q = row >> 1;
        int tt = q % NTP; q /= NTP;
        int yy = q % NXP;
        int xx = q / NXP;
        int d  = 16*jb + m;
        int pa = d >> 4, pb = (d >> 2) & 3, pc = d & 3;
        atomicAdd(&acc[((cch*ACCX + xx+pa)*ACCX + yy+pb)*ACCT + tt+pc], val);
      }
    }
  }
}

// ------------------------------------------------------------------
extern "C" void kernel_launch(void* const* d_in, const int* in_sizes, int n_in,
                              void* d_out, int out_size, void* d_ws, size_t ws_size,
                              hipStream_t stream){
  (void)in_sizes; (void)n_in; (void)out_size; (void)ws_size;
  const float* y_ri = (const float*)d_in[0];
  const float* dico = (const float*)d_in[1];
  const float* lamr = (const float*)d_in[2];
  const float* betr = (const float*)d_in[3];
  const int*   mask = (const int*)d_in[4];
  float* ws  = (float*)d_ws;
  float* out = (float*)d_out;

  float* xre = ws+WS_XRE; float* xim = ws+WS_XIM;
  float* yre = ws+WS_YRE; float* yim = ws+WS_YIM;
  float* bre = ws+WS_BRE; float* bim = ws+WS_BIM;
  float* cre = ws+WS_CRE; float* cim = ws+WS_CIM;
  float* twc = ws+WS_TWC; float* tws = ws+WS_TWS;

  k_init <<<(NVOX+255)/256, 256, 0, stream>>>(y_ri, ws);
  k_dtd  <<<64, 256, 0, stream>>>(dico, ws);
  k_power<<<1, 128, 0, stream>>>(dico, lamr, betr, ws);

  // x = ifft2(y) (ortho)
  k_dft<<<960, 80, 0, stream>>>(yre, yim, bre, bim, twc, tws, 0,  1.0f);
  k_dft<<<960, 80, 0, stream>>>(bre, bim, xre, xim, twc, tws, 1,  1.0f);

  const size_t fista_lds = (size_t)LDS_FLOATS * sizeof(float);   // 133120 B
  for (int o = 0; o < 2; ++o){
    k_zero <<<(NACC+255)/256, 256, 0, stream>>>(ws+WS_ACC, NACC);
    k_fista<<<(TILES+3)/4, 128, fista_lds, stream>>>(xre, xim, dico, ws+WS_DTD,
                                                     ws+WS_COLS, ws+WS_SCAL, ws+WS_ACC);
    k_crop <<<(2*NVOX+255)/256, 256, 0, stream>>>(ws+WS_ACC, cre, cim);
    // data consistency: fft2 -> blend -> ifft2
    k_dft  <<<960, 80, 0, stream>>>(cre, cim, bre, bim, twc, tws, 0, -1.0f);
    k_dft  <<<960, 80, 0, stream>>>(bre, bim, cre, cim, twc, tws, 1, -1.0f);
    k_blend<<<(NVOX+255)/256, 256, 0, stream>>>(cre, cim, yre, yim, mask, ws+WS_SCAL);
    k_dft  <<<960, 80, 0, stream>>>(cre, cim, bre, bim, twc, tws, 0,  1.0f);
    k_dft  <<<960, 80, 0, stream>>>(bre, bim, xre, xim, twc, tws, 1,  1.0f);
  }

  k_out<<<(NVOX+255)/256, 256, 0, stream>>>(xre, xim, out);
}